// LSTMSoftAttention_67156108640578
// MI455X (gfx1250) — compile-verified
//
#include <hip/hip_runtime.h>
#include <hip/hip_bf16.h>

#define H  1024
#define NN 2048
#define T  2048
#define B  16

typedef float v2f __attribute__((ext_vector_type(2)));
typedef float v8f __attribute__((ext_vector_type(8)));

// tanh(x) = 1 - 2/(exp(2x)+1)  -- stable at +-inf (no inf/inf), ~1ulp HW trans ops
__device__ __forceinline__ float fast_tanh(float x) {
    float e = __builtin_amdgcn_exp2f(x * 2.8853900817779268f); // exp(2x)
    return 1.0f - 2.0f * __builtin_amdgcn_rcpf(e + 1.0f);
}

// ---------------- Phase A: dec_fea = dec @ Wd.T + bd via V_WMMA_F32_16X16X4_F32 ----
// One wave per 16x16 output tile (M=16 rows of dec x 16 cols n0..n0+15), K-loop of 256 WMMAs.
// A frag (16x4 f32): lanes 0-15 -> M=lane, {v0,v1}={K,K+1}; lanes 16-31 -> {K+2,K+3}.
// B frag (4x16 f32): lane&15 -> N, same K grouping (B[k][n] = Wd[n0+n, kbase+k]).
// D frag: VGPR r holds row r (lanes 0-15) / row r+8 (lanes 16-31), col = lane&15.
__global__ __launch_bounds__(256) void decfea_wmma(const float* __restrict__ dec,
                                                   const float* __restrict__ Wd,
                                                   const float* __restrict__ bd,
                                                   float* __restrict__ decfea) {
    int wave = threadIdx.x >> 5;
    int lane = threadIdx.x & 31;
    int nt = blockIdx.x * 8 + wave;           // 0..127 n-tiles
    int n0 = nt * 16;
    int m  = lane & 15;
    int ks = (lane >> 4) << 1;                // 0 or 2
    const float* ap = dec + (size_t)m * H + ks;
    const float* bp = Wd + (size_t)(n0 + m) * H + ks;
    v8f c = {0.f, 0.f, 0.f, 0.f, 0.f, 0.f, 0.f, 0.f};
    for (int k = 0; k < H; k += 4) {
        v2f a = *(const v2f*)(ap + k);
        v2f b = *(const v2f*)(bp + k);
        c = __builtin_amdgcn_wmma_f32_16x16x4_f32(false, a, false, b,
                                                  (short)0, c, false, false);
    }
    int col   = lane & 15;
    int rbase = (lane >> 4) << 3;
    float bias = bd[n0 + col];
    #pragma unroll
    for (int r = 0; r < 8; ++r)
        decfea[(size_t)(rbase + r) * NN + n0 + col] = c[r] + bias;
}

// ---------------- Phase B: scores[t,b] = sum_n tanh(enc+decfea+cov*wc) * v ---------
// One wave per (t,b); contiguous 8KB row, float4 loads, wave shuffle reduction.
__global__ __launch_bounds__(256) void scores_kernel(const float* __restrict__ enc,
                                                     const float* __restrict__ decfea,
                                                     const float* __restrict__ cov,
                                                     const float* __restrict__ wc,
                                                     const float* __restrict__ vv,
                                                     float* __restrict__ scores) {
    int gw   = (blockIdx.x * 256 + threadIdx.x) >> 5;  // global wave 0..32767
    int lane = threadIdx.x & 31;
    int t = gw >> 4;
    int b = gw & 15;
    size_t base = (size_t)(t * B + b) * NN;
    float cval = cov[t * B + b];
    const float4* e4 = (const float4*)(enc + base);
    const float4* d4 = (const float4*)(decfea + (size_t)b * NN);
    const float4* w4 = (const float4*)wc;
    const float4* v4 = (const float4*)vv;
    float acc = 0.f;
    #pragma unroll 4
    for (int i = lane; i < NN / 4; i += 32) {
        float4 e = e4[i]; float4 d = d4[i]; float4 w = w4[i]; float4 vf = v4[i];
        acc += fast_tanh(e.x + d.x + cval * w.x) * vf.x;
        acc += fast_tanh(e.y + d.y + cval * w.y) * vf.y;
        acc += fast_tanh(e.z + d.z + cval * w.z) * vf.z;
        acc += fast_tanh(e.w + d.w + cval * w.w) * vf.w;
    }
    #pragma unroll
    for (int off = 16; off > 0; off >>= 1) acc += __shfl_xor(acc, off, 32);
    if (lane == 0) scores[t * B + b] = acc;
}

// ---------------- Phase C: masked softmax over T per column b ----------------------
__global__ __launch_bounds__(256) void softmax_kernel(const float* __restrict__ scores,
                                                      const int* __restrict__ mask,
                                                      const float* __restrict__ cov,
                                                      float* __restrict__ attn,
                                                      float* __restrict__ covout) {
    int b = blockIdx.x;
    int tid = threadIdx.x;
    __shared__ float sred[8];
    float loc[8];
    float mx = -__builtin_inff();
    #pragma unroll
    for (int j = 0; j < 8; ++j) {
        int t = tid + j * 256;
        float s = scores[t * B + b];
        if (mask[t * B + b] != 0) s = -__builtin_inff();
        loc[j] = s;
        mx = fmaxf(mx, s);
    }
    #pragma unroll
    for (int off = 16; off > 0; off >>= 1) mx = fmaxf(mx, __shfl_xor(mx, off, 32));
    if ((tid & 31) == 0) sred[tid >> 5] = mx;
    __syncthreads();
    mx = sred[0];
    #pragma unroll
    for (int w = 1; w < 8; ++w) mx = fmaxf(mx, sred[w]);
    __syncthreads();
    float sum = 0.f;
    #pragma unroll
    for (int j = 0; j < 8; ++j) {
        float e = __builtin_amdgcn_exp2f((loc[j] - mx) * 1.4426950408889634f);
        loc[j] = e;                         // exp(-inf)=0 for masked entries
        sum += e;
    }
    #pragma unroll
    for (int off = 16; off > 0; off >>= 1) sum += __shfl_xor(sum, off, 32);
    if ((tid & 31) == 0) sred[tid >> 5] = sum;
    __syncthreads();
    sum = 0.f;
    #pragma unroll
    for (int w = 0; w < 8; ++w) sum += sred[w];
    float inv = 1.0f / sum;
    #pragma unroll
    for (int j = 0; j < 8; ++j) {
        int t = tid + j * 256;
        float a = loc[j] * inv;
        attn[t * B + b]   = a;
        covout[t * B + b] = cov[t * B + b] + a;
    }
}

// ---------------- Phase D: partial context sums over t-chunks ----------------------
__global__ __launch_bounds__(256) void context_partial(const float* __restrict__ enc,
                                                       const float* __restrict__ attn,
                                                       float* __restrict__ part,
                                                       int tpc) {
    int n  = blockIdx.x * 256 + threadIdx.x;   // 0..2047
    int b  = blockIdx.y;                       // 0..15
    int tc = blockIdx.z;                       // 0..tch-1
    int t0 = tc * tpc;
    float acc = 0.f;
    #pragma unroll 4
    for (int tt = 0; tt < tpc; ++tt) {
        int tb = (t0 + tt) * B + b;
        float a = attn[tb];                    // block-uniform -> scalar load
        acc += a * enc[(size_t)tb * NN + n];
    }
    part[((size_t)tc * B + b) * NN + n] = acc;
}

// ---------------- Phase E: reduce partials into c[b,n] -----------------------------
__global__ __launch_bounds__(256) void context_reduce(const float* __restrict__ part,
                                                      float* __restrict__ cout,
                                                      int tch) {
    int idx = blockIdx.x * 256 + threadIdx.x;  // = b*NN + n, 0..32767
    float acc = 0.f;
    for (int tc = 0; tc < tch; ++tc)
        acc += part[(size_t)tc * (B * NN) + idx];
    cout[idx] = acc;
}

extern "C" void kernel_launch(void* const* d_in, const int* in_sizes, int n_in,
                              void* d_out, int out_size, void* d_ws, size_t ws_size,
                              hipStream_t stream) {
    const float* dec  = (const float*)d_in[0];   // [B, H]
    const float* enc  = (const float*)d_in[1];   // [T, B, N]
    const int*   mask = (const int*)d_in[2];     // [T, B]
    const float* cov  = (const float*)d_in[3];   // [T, B]
    const float* Wd   = (const float*)d_in[4];   // [N, H]
    const float* bd   = (const float*)d_in[5];   // [N]
    const float* wc   = (const float*)d_in[6];   // [N]
    const float* vv   = (const float*)d_in[7];   // [N]

    float* out      = (float*)d_out;
    float* c_out    = out;                       // [B, N]  32768
    float* attn_out = out + B * NN;              // [T, B]  32768
    float* cov_out  = out + B * NN + T * B;      // [T, B]  32768

    float* ws     = (float*)d_ws;
    float* decfea = ws;                          // 32768 floats
    float* scores = ws + B * NN;                 // 32768 floats
    float* part   = ws + 2 * B * NN;             // tch * 32768 floats

    // Pick largest power-of-two t-chunk count that fits in workspace (parallelism
    // for the bandwidth-bound context pass); tch=64 -> 8192 blocks, 8.25 MiB ws.
    size_t avail = (ws_size / sizeof(float) > (size_t)(2 * B * NN))
                   ? ws_size / sizeof(float) - 2 * B * NN : 0;
    int tch = 1;
    while (tch < 64 && (size_t)(tch * 2) * (B * NN) <= avail) tch *= 2;
    int tpc = T / tch;

    decfea_wmma   <<<16,               256, 0, stream>>>(dec, Wd, bd, decfea);
    scores_kernel <<<(T * B) / 8,      256, 0, stream>>>(enc, decfea, cov, wc, vv, scores);
    softmax_kernel<<<B,                256, 0, stream>>>(scores, mask, cov, attn_out, cov_out);
    context_partial<<<dim3(NN / 256, B, tch), 256, 0, stream>>>(enc, attn_out, part, tpc);
    context_reduce <<<(B * NN) / 256,  256, 0, stream>>>(part, c_out, tch);
}